// ToroIndexer_5677946765984
// MI455X (gfx1250) — compile-verified
//
#include <hip/hip_runtime.h>
#include <hip/hip_bf16.h>

typedef float v2f __attribute__((ext_vector_type(2)));
typedef float v8f __attribute__((ext_vector_type(8)));

#define DIST  0.2f
#define NTOP  200
#define ARES  90
#define NB    8
#define NPK   80
#define NM    (NTOP*ARES)
#define TPW   4        // direction tiles per wave in phase 1

// ---------------- small 3x3 helpers ----------------
__device__ __forceinline__ void norm3(const float v[3], float o[3]) {
  float n = sqrtf(v[0]*v[0]+v[1]*v[1]+v[2]*v[2]) + 1e-12f;
  o[0]=v[0]/n; o[1]=v[1]/n; o[2]=v[2]/n;
}
__device__ __forceinline__ void mm3(const float A[3][3], const float B[3][3], float C[3][3]) {
  for (int i=0;i<3;i++) for (int j=0;j<3;j++)
    C[i][j]=A[i][0]*B[0][j]+A[i][1]*B[1][j]+A[i][2]*B[2][j];
}
// adjugate/det inverse, singular -> zero (matches reference)
__device__ __forceinline__ void inv3x3(const float M[3][3], float R[3][3]) {
  float a=M[0][0],b=M[0][1],c=M[0][2];
  float d=M[1][0],e=M[1][1],f=M[1][2];
  float g=M[2][0],h=M[2][1],i=M[2][2];
  float det = a*(e*i-f*h) - b*(d*i-f*g) + c*(d*h-e*g);
  float invdet = (fabsf(det) >= 1e-10f) ? (1.0f/det) : 0.0f;
  R[0][0]=(e*i-f*h)*invdet; R[0][1]=(c*h-b*i)*invdet; R[0][2]=(b*f-c*e)*invdet;
  R[1][0]=(f*g-d*i)*invdet; R[1][1]=(a*i-c*g)*invdet; R[1][2]=(c*d-a*f)*invdet;
  R[2][0]=(d*h-e*g)*invdet; R[2][1]=(b*g-a*h)*invdet; R[2][2]=(a*e-b*d)*invdet;
}
__device__ __forceinline__ void anglesDeg(const float M[3][3], float out[3]) {
  float n[3][3];
  for (int r=0;r<3;r++) norm3(M[r], n[r]);
  const float TODEG = 57.295779513082320877f;
  float d01=n[0][0]*n[1][0]+n[0][1]*n[1][1]+n[0][2]*n[1][2];
  float d02=n[0][0]*n[2][0]+n[0][1]*n[2][1]+n[0][2]*n[2][2];
  float d12=n[1][0]*n[2][0]+n[1][1]*n[2][1]+n[1][2]*n[2][2];
  d01=fminf(1.f,fmaxf(-1.f,d01));
  d02=fminf(1.f,fmaxf(-1.f,d02));
  d12=fminf(1.f,fmaxf(-1.f,d12));
  out[0]=acosf(d01)*TODEG; out[1]=acosf(d02)*TODEG; out[2]=acosf(d12)*TODEG;
}
// R0(align cell0->dir) then Rodrigues roll around dir: B = Rr * R0 * cell_rows
__device__ __forceinline__ void buildCandidate(const float cell[3][3], const float dir[3],
                                               float alpha, float B[3][3]) {
  float R0[3][3];
  {
    float sn[3], tn[3];
    norm3(cell[0], sn); norm3(dir, tn);
    float V0=sn[1]*tn[2]-sn[2]*tn[1], V1=sn[2]*tn[0]-sn[0]*tn[2], V2=sn[0]*tn[1]-sn[1]*tn[0];
    float C =sn[0]*tn[0]+sn[1]*tn[1]+sn[2]*tn[2];
    float S2=V0*V0+V1*V1+V2*V2;
    float K[3][3]={{0.f,-V2,V1},{V2,0.f,-V0},{-V1,V0,0.f}};
    float KK[3][3]; mm3(K,K,KK);
    float fac=(1.0f-C)/(S2+1e-12f);
    for (int i=0;i<3;i++) for (int j=0;j<3;j++)
      R0[i][j]=((i==j)?1.f:0.f)+K[i][j]+fac*KK[i][j];
  }
  float cells[3][3];
  for (int r=0;r<3;r++) for (int i=0;i<3;i++)
    cells[r][i]=R0[i][0]*cell[r][0]+R0[i][1]*cell[r][1]+R0[i][2]*cell[r][2];
  float Rr[3][3];
  {
    float an[3]; norm3(dir,an);
    float A[3][3]={{0.f,-an[2],an[1]},{an[2],0.f,-an[0]},{-an[1],an[0],0.f}};
    float AA[3][3]; mm3(A,A,AA);
    float s=sinf(alpha), c=1.0f-cosf(alpha);
    for (int i=0;i<3;i++) for (int j=0;j<3;j++)
      Rr[i][j]=((i==j)?1.f:0.f)+s*A[i][j]+c*AA[i][j];
  }
  for (int r=0;r<3;r++) for (int i=0;i<3;i++)
    B[r][i]=Rr[i][0]*cells[r][0]+Rr[i][1]*cells[r][1]+Rr[i][2]*cells[r][2];
}

// ---------------- Phase 1: WMMA f32 16x16x4 direction scoring ----------------
// Block = 256 threads = 8 waves; wave w = batch b. Each wave keeps 5 A tiles
// (80 peaks x K4) in VGPRs and sweeps TPW direction tiles (B operands),
// doing 5*TPW v_wmma_f32_16x16x4_f32. Operand selects are branch-free
// (register cndmask, no exec-mask load divergence).
__global__ __launch_bounds__(256) void k_score(const float* __restrict__ peaks,
                                               const float* __restrict__ cell,
                                               const float* __restrict__ sph,
                                               int L, int* __restrict__ score) {
  int b    = threadIdx.x >> 5;     // wave-in-block == batch
  int lane = threadIdx.x & 31;
  int col  = lane & 15;
  bool hi  = (lane & 16) != 0;

  float c0 = cell[0], c1 = cell[1], c2 = cell[2];
  float inv_s = 1.0f / sqrtf(c0*c0 + c1*c1 + c2*c2);

  // Load the 5 A tiles once: lanes 0-15 hold {x,y} (K0,K1), lanes 16-31 {z,0} (K2,K3).
  v2f av[5];
  #pragma unroll
  for (int pt = 0; pt < 5; ++pt) {
    const float* pk = peaks + ((size_t)(b*NPK) + pt*16 + col)*3;
    float x = pk[0], y = pk[1], z = pk[2];   // unconditional b96-mergeable loads
    av[pt].x = hi ? z : x;
    av[pt].y = hi ? 0.0f : y;
  }

  int tile0 = blockIdx.x * TPW;
  #pragma unroll
  for (int tt = 0; tt < TPW; ++tt) {
    int j  = (tile0 + tt)*16 + col;
    int jc = (j < L) ? j : (L-1);
    const float* dp = sph + (size_t)3*jc;
    float dx = dp[0], dy = dp[1], dz = dp[2];
    v2f bv; bv.x = hi ? dz : dx; bv.y = hi ? 0.0f : dy;

    int cnt = 0;
#if defined(__gfx1250__) && __has_builtin(__builtin_amdgcn_wmma_f32_16x16x4_f32)
    #pragma unroll
    for (int pt = 0; pt < 5; ++pt) {
      v8f c = {0.f,0.f,0.f,0.f,0.f,0.f,0.f,0.f};
      v8f d = __builtin_amdgcn_wmma_f32_16x16x4_f32(
          false, av[pt], false, bv, (short)0, c, false, false);
      #pragma unroll
      for (int v = 0; v < 8; ++v) {
        float f = d[v] * inv_s;
        f = fabsf(f - rintf(f));
        cnt += (f < DIST) ? 1 : 0;
      }
    }
#else
    // scalar fallback (host pass / missing builtin): half==0 lanes own dir j fully
    if (!hi) {
      for (int pt = 0; pt < 5; ++pt)
        for (int r = 0; r < 16; ++r) {
          const float* q = peaks + ((size_t)(b*NPK) + pt*16 + r)*3;
          float f = (q[0]*dx + q[1]*dy + q[2]*dz) * inv_s;
          f = fabsf(f - rintf(f));
          cnt += (f < DIST) ? 1 : 0;
        }
    }
#endif
    cnt += __shfl_xor(cnt, 16, 32);   // fold D halves (M 0-7 + M 8-15)
    if (!hi && j < L) score[(size_t)b*L + j] = cnt;
  }
}

// ---------------- Phase 1b: deterministic top-200 (histogram + ordered scan) ------
// Wave32 shfl-based block scan: 3 barriers per chunk instead of 20.
__global__ __launch_bounds__(1024) void k_top(const int* __restrict__ score, int L,
                                              int* __restrict__ topIdx) {
  int b = blockIdx.x;
  int tid  = threadIdx.x;
  int lane = tid & 31;
  int wid  = tid >> 5;
  __shared__ int hist[81];
  __shared__ int wtot[32];
  __shared__ int sT[2];
  if (tid < 81) hist[tid] = 0;
  __syncthreads();
  for (int i = tid; i < L; i += 1024) {
    int s = score[(size_t)b*L + i];
    s = (s < 0) ? 0 : ((s > 80) ? 80 : s);
    atomicAdd(&hist[s], 1);
  }
  __syncthreads();
  if (tid == 0) {
    int cum = 0, T = 0, nAb = 0;
    for (int s = 80; s >= 0; --s) {
      if (cum + hist[s] >= NTOP || s == 0) { T = s; nAb = cum; break; }
      cum += hist[s];
    }
    sT[0] = T; sT[1] = nAb;
  }
  __syncthreads();
  int T = sT[0];
  int c1 = 0, c2 = sT[1];
  int nChunk = (L + 1023) / 1024;
  for (int ch = 0; ch < nChunk; ++ch) {
    int i = ch*1024 + tid;
    int s = (i < L) ? score[(size_t)b*L + i] : -1;
    int p1 = (s >  T) ? 1 : 0;
    int p2 = (s == T) ? 1 : 0;
    int incl = p1 | (p2 << 16);        // packed dual scan (counts < 2^16)
    #pragma unroll
    for (int off = 1; off < 32; off <<= 1) {
      int t = __shfl_up(incl, off, 32);
      if (lane >= off) incl += t;
    }
    if (lane == 31) wtot[wid] = incl;
    __syncthreads();
    if (wid == 0) {
      int wv = wtot[lane];
      #pragma unroll
      for (int off = 1; off < 32; off <<= 1) {
        int t = __shfl_up(wv, off, 32);
        if (lane >= off) wv += t;
      }
      wtot[lane] = wv;
    }
    __syncthreads();
    int base = (wid > 0) ? wtot[wid-1] : 0;
    incl += base;
    int tot = wtot[31];
    int e1 = (incl & 0xFFFF) - p1;
    int e2 = (incl >> 16)    - p2;
    if (p1) { int slot = c1 + e1; if (slot < NTOP) topIdx[b*NTOP + slot] = i; }
    if (p2) { int slot = c2 + e2; if (slot < NTOP) topIdx[b*NTOP + slot] = i; }
    c1 += tot & 0xFFFF;
    c2 += tot >> 16;
    __syncthreads();                  // protect wtot before next chunk
  }
}

// ---------------- Phase 2: 18000 candidate cells/batch, count indexed peaks -------
__global__ __launch_bounds__(256) void k_cand(const float* __restrict__ peaks,
                                              const float* __restrict__ cellG,
                                              const float* __restrict__ sph,
                                              const int* __restrict__ topIdx,
                                              int* __restrict__ cscore) {
  int b = blockIdx.y;
  int m = blockIdx.x*256 + threadIdx.x;
  __shared__ float sp[NPK*3];
  if (threadIdx.x < NPK*3) sp[threadIdx.x] = peaks[(size_t)b*NPK*3 + threadIdx.x];
  __syncthreads();
  if (m >= NM) return;
  float cell[3][3];
  for (int r=0;r<3;r++) for (int i=0;i<3;i++) cell[r][i]=cellG[r*3+i];
  int k = m / ARES, a = m % ARES;
  int di = topIdx[b*NTOP + k];
  float dir[3] = { sph[3*di], sph[3*di+1], sph[3*di+2] };
  float alpha = 6.2831853071795864769f * (float)a / (float)ARES;
  float B[3][3]; buildCandidate(cell, dir, alpha, B);
  float inv[3][3]; inv3x3(B, inv);
  int cnt = 0;
  for (int n = 0; n < NPK; ++n) {
    float px=sp[3*n], py=sp[3*n+1], pz=sp[3*n+2];
    float emax = 0.f;
    #pragma unroll
    for (int kk=0; kk<3; ++kk) {
      float h = px*inv[0][kk] + py*inv[1][kk] + pz*inv[2][kk];
      float e = fabsf(h - rintf(h));
      emax = fmaxf(emax, e);
    }
    cnt += (emax < DIST) ? 1 : 0;
  }
  cscore[(size_t)b*NM + m] = cnt;
}

// ---------------- Phase 2b: first-max argmax per batch ----------------------------
__global__ __launch_bounds__(256) void k_argmax(const int* __restrict__ cscore,
                                                int* __restrict__ bestm,
                                                float* __restrict__ dout) {
  int b = blockIdx.x, tid = threadIdx.x;
  __shared__ int ss[256], si[256];
  int bs = -1, bi = NM;
  for (int m = tid; m < NM; m += 256) {
    int s = cscore[(size_t)b*NM + m];
    if (s > bs) { bs = s; bi = m; }     // ascending stride -> first-max per thread
  }
  ss[tid] = bs; si[tid] = bi;
  __syncthreads();
  for (int off = 128; off > 0; off >>= 1) {
    if (tid < off) {
      int s2 = ss[tid+off], i2 = si[tid+off];
      if (s2 > ss[tid] || (s2 == ss[tid] && i2 < si[tid])) { ss[tid]=s2; si[tid]=i2; }
    }
    __syncthreads();
  }
  if (tid == 0) { bestm[b] = si[0]; dout[72 + b] = (float)ss[0]; }
}

// ---------------- Phase 3: one wave per batch, 5x weighted LSQ + penalization -----
__global__ __launch_bounds__(32) void k_refine(const float* __restrict__ peaks,
                                               const float* __restrict__ cellG,
                                               const float* __restrict__ sph,
                                               const int* __restrict__ topIdx,
                                               const int* __restrict__ bestm,
                                               float* __restrict__ dout) {
  int b = blockIdx.x;
  int lane = threadIdx.x;
  float cell[3][3];
  for (int r=0;r<3;r++) for (int i=0;i<3;i++) cell[r][i]=cellG[r*3+i];
  int bm = bestm[b];
  int k = bm / ARES, a = bm % ARES;
  int di = topIdx[b*NTOP + k];
  float dir[3] = { sph[3*di], sph[3*di+1], sph[3*di+2] };
  float alpha = 6.2831853071795864769f * (float)a / (float)ARES;
  float B[3][3]; buildCandidate(cell, dir, alpha, B);   // identical across lanes
  const float* pk = peaks + (size_t)b*NPK*3;

  for (int t = 0; t < 5; ++t) {
    float thr = 0.02f + (float)t * ((1.5e-3f - 0.02f) * 0.25f);  // linspace(0.02,1.5e-3,5)
    float inv[3][3]; inv3x3(B, inv);
    float acc[18];
    #pragma unroll
    for (int q=0;q<18;q++) acc[q]=0.f;
    for (int n = lane; n < NPK; n += 32) {
      float p[3] = { pk[3*n], pk[3*n+1], pk[3*n+2] };
      float h[3];
      #pragma unroll
      for (int kk=0; kk<3; ++kk)
        h[kk] = rintf(p[0]*inv[0][kk] + p[1]*inv[1][kk] + p[2]*inv[2][kk]);
      float rr = 0.f;
      #pragma unroll
      for (int d=0; d<3; ++d) {
        float rv = h[0]*B[0][d] + h[1]*B[1][d] + h[2]*B[2][d] - p[d];
        rr += rv*rv;
      }
      if (sqrtf(rr) < thr) {
        #pragma unroll
        for (int kk=0; kk<3; ++kk)
          #pragma unroll
          for (int l=0; l<3; ++l) acc[kk*3+l] += h[kk]*h[l];
        #pragma unroll
        for (int kk=0; kk<3; ++kk)
          #pragma unroll
          for (int d=0; d<3; ++d) acc[9+kk*3+d] += h[kk]*p[d];
      }
    }
    for (int off = 16; off > 0; off >>= 1)
      #pragma unroll
      for (int q=0; q<18; q++) acc[q] += __shfl_xor(acc[q], off, 32);
    float XtX[3][3], XtY[3][3];
    for (int kk=0;kk<3;kk++) for (int l=0;l<3;l++) { XtX[kk][l]=acc[kk*3+l]; XtY[kk][l]=acc[9+kk*3+l]; }
    float iX[3][3]; inv3x3(XtX, iX);
    mm3(iX, XtY, B);
  }

  if (lane == 0) {
    for (int r=0;r<3;r++) for (int i=0;i<3;i++) dout[b*9 + r*3 + i] = B[r][i];
    float angB[3], angC[3];
    anglesDeg(B, angB);
    anglesDeg(cell, angC);
    float dc = 0.f, da = 0.f;
    for (int r=0;r<3;r++) {
      float nb = sqrtf(B[r][0]*B[r][0]+B[r][1]*B[r][1]+B[r][2]*B[r][2]);
      float nc = sqrtf(cell[r][0]*cell[r][0]+cell[r][1]*cell[r][1]+cell[r][2]*cell[r][2]);
      dc = fmaxf(dc, fabsf(nb - nc));
    }
    for (int q=0;q<3;q++) da = fmaxf(da, fabsf(angB[q]-angC[q]));
    float pen = dc*dc*dc + da*da*da;   // max(x,0)^3 with x>=0
    dout[80 + b] = pen;
  }
}

// ---------------- launcher --------------------------------------------------------
extern "C" void kernel_launch(void* const* d_in, const int* in_sizes, int n_in,
                              void* d_out, int out_size, void* d_ws, size_t ws_size,
                              hipStream_t stream) {
  const float* peaks = (const float*)d_in[0];   // (8,80,3)
  const float* cell  = (const float*)d_in[1];   // (3,3)
  const float* sph   = (const float*)d_in[2];   // (L,3)
  int L = in_sizes[2] / 3;                      // 100000

  char* ws   = (char*)d_ws;
  int* score  = (int*)ws;                                              // 8*L ints
  int* topIdx = (int*)(ws + (size_t)NB * L * sizeof(int));             // 8*200 ints
  int* cscore = (int*)((char*)topIdx + (size_t)NB*NTOP*sizeof(int));   // 8*18000 ints
  int* bestm  = (int*)((char*)cscore + (size_t)NB*NM*sizeof(int));     // 8 ints
  float* out  = (float*)d_out;                  // [B(72) | best_count(8) | pen(8)]

  int nTiles  = (L + 15) / 16;
  int nBlocks = (nTiles + TPW - 1) / TPW;
  k_score <<<nBlocks, 256, 0, stream>>>(peaks, cell, sph, L, score);
  k_top   <<<NB, 1024, 0, stream>>>(score, L, topIdx);
  dim3 g3((NM + 255)/256, NB);
  k_cand  <<<g3, 256, 0, stream>>>(peaks, cell, sph, topIdx, cscore);
  k_argmax<<<NB, 256, 0, stream>>>(cscore, bestm, out);
  k_refine<<<NB, 32, 0, stream>>>(peaks, cell, sph, topIdx, bestm, out);
}